// ContrastiveLoss_16192026706341
// MI455X (gfx1250) — compile-verified
//
#include <hip/hip_runtime.h>
#include <math.h>

// Problem constants (from reference)
#define S_DIM   32
#define T_DIM   512
#define F_DIM   512
#define STEPS   12
#define NEG     32

// Tiling
#define U_TILE  4                    // consecutive u per block (1 wave per u)
#define NTHREADS (32 * U_TILE)       // 128
#define KC      32                   // K-chunk in floats
#define NCHUNK  (F_DIM / KC)         // 16
#define AROWS   (U_TILE + 11)        // 15 prediction time-rows staged per block
#define AF4     (AROWS * (KC * STEPS / 4))   // 1440 float4 per A chunk
#define BROWS   33                   // 1 positive + 32 negative target rows per u
#define BSTRIDE (KC + 4)             // 36: rows 16B-aligned AND bank-conflict-free

typedef __attribute__((ext_vector_type(2))) float v2f;
typedef __attribute__((ext_vector_type(8))) float v8f;

// Low 32 bits of a generic pointer to LDS == the DS-relative byte offset.
__device__ __forceinline__ unsigned lds_off(const void* p) {
    return (unsigned)(uintptr_t)p;
}
// CDNA5 async copy global->LDS, 16B per lane, tracked by ASYNCcnt.
__device__ __forceinline__ void async_load_b128(unsigned lds, const float* g) {
    asm volatile("global_load_async_to_lds_b128 %0, %1, off"
                 :: "v"(lds), "v"((unsigned long long)(uintptr_t)g)
                 : "memory");
}
__device__ __forceinline__ void wait_async0() {
    asm volatile("s_wait_asynccnt 0x0" ::: "memory");
}
// ds_swizzle_b32 xor-shuffle: offset = {0, xor[14:10], or[9:5]=0, and[4:0]=0x1F}
#define SWZ(x, imm) \
    __int_as_float(__builtin_amdgcn_ds_swizzle(__float_as_int(x), (imm)))

// One wave computes the 16x48x512 logits GEMM for one (s,u) with
// v_wmma_f32_16x16x4_f32, operands staged in LDS via async-to-LDS copies.
__global__ __launch_bounds__(NTHREADS) void cpc_wmma_loss_kernel(
    const float* __restrict__ true_latent,   // (S, T, F)
    const float* __restrict__ predictions,   // (S, T, F, STEPS)
    const int*   __restrict__ neg_indices,   // (S, NEG*T)
    float*       __restrict__ partial)       // (S*T) per-(s,u) loss
{
    __shared__ float sA[AROWS * KC * STEPS];        // 15*384  = 5760 floats
    __shared__ float sB[U_TILE * BROWS * BSTRIDE];  // 4*33*36 = 4752 floats

    const int tid  = threadIdx.x;
    const int w    = tid >> 5;                 // wave id = u sub-index
    const int lane = tid & 31;
    const int s    = blockIdx.x / (T_DIM / U_TILE);
    const int u0   = (blockIdx.x % (T_DIM / U_TILE)) * U_TILE;
    const int u    = u0 + w;
    const int h    = lane >> 4;                // K-half (ISA A/B frag layout)
    const int n    = lane & 15;                // M row (A) / N col (B)

    // ---- per-lane compute-fragment bases (float indices into LDS) ----
    // A row m = n (step); t = u - m => staged row trow = w + 11 - m_eff.
    const int m_eff = (n < STEPS) ? n : (STEPS - 1);
    const int aBase = (w + 11 - m_eff) * (KC * STEPS) + 24 * h + m_eff;
    const int b0Base = (w * BROWS +  n      ) * BSTRIDE + 2 * h;  // c = 0..15
    const int b1Base = (w * BROWS + 16 + n  ) * BSTRIDE + 2 * h;  // c = 16..31
    const int b2Base = (w * BROWS + ((n == 0) ? 32 : 1)) * BSTRIDE + 2 * h;

    // ---- B staging lane mapping: 4 rows x 8 float4 per iteration ----
    const int sf4  = lane & 7;
    const int rsub = lane >> 3;

    const size_t predS = (size_t)s * T_DIM * (F_DIM * STEPS);
    const size_t latS  = (size_t)s * T_DIM * F_DIM;
    const int tbase = u0 - 11;

    v8f acc0 = {}, acc1 = {}, acc2 = {};

    for (int chunk = 0; chunk < NCHUNK; ++chunk) {
        const int kc12 = chunk * (KC * STEPS);   // float offset in (k,m) block

        // ---- stage A cooperatively: 15 rows x 1536B contiguous, coalesced ----
        for (int idx = tid; idx < AF4; idx += NTHREADS) {
            const int tr  = idx / (KC * STEPS / 4);
            const int off = idx - tr * (KC * STEPS / 4);
            int t = tbase + tr; if (t < 0) t = 0;      // t<0 rows masked later
            const float* g = predictions + predS +
                             (size_t)t * (F_DIM * STEPS) + kc12 + off * 4;
            async_load_b128(lds_off(&sA[tr * (KC * STEPS) + off * 4]), g);
        }
        // ---- stage B: each wave gathers its own 33 target row-chunks ----
        for (int it = 0; it < 9; ++it) {
            const int r = it * 4 + rsub;               // r == copy index c
            if (r < BROWS) {
                int tau;
                if (r == 0) {
                    tau = u;                           // positive sample
                } else {
                    // reference quirk preserved: skip test uses j/NEG
                    const int j    = (r - 1) * T_DIM + u;
                    const int idx0 = neg_indices[(size_t)s * (NEG * T_DIM) + j];
                    tau = idx0 + ((idx0 >= (j >> 5)) ? 1 : 0);
                }
                const float* g = true_latent + latS +
                                 (size_t)tau * F_DIM + chunk * KC + sf4 * 4;
                async_load_b128(
                    lds_off(&sB[(w * BROWS + r) * BSTRIDE + sf4 * 4]), g);
            }
        }
        wait_async0();          // my async copies landed in LDS
        __syncthreads();        // everyone's copies landed

        // ---- 8 K-steps of fp32 WMMA x 3 N-tiles (all operands from LDS) ----
        #pragma unroll
        for (int kk = 0; kk < KC / 4; ++kk) {
            v2f a, b0, b1, b2;
            a[0] = sA[aBase + kk * 48];        // k   = 4kk + 2h
            a[1] = sA[aBase + kk * 48 + 12];   // k+1
            const float2 q0 = *(const float2*)&sB[b0Base + 4 * kk];
            const float2 q1 = *(const float2*)&sB[b1Base + 4 * kk];
            const float2 q2 = *(const float2*)&sB[b2Base + 4 * kk];
            b0[0] = q0.x; b0[1] = q0.y;
            b1[0] = q1.x; b1[1] = q1.y;
            b2[0] = q2.x; b2[1] = q2.y;
            acc0 = __builtin_amdgcn_wmma_f32_16x16x4_f32(
                       false, a, false, b0, (short)0, acc0, false, false);
            acc1 = __builtin_amdgcn_wmma_f32_16x16x4_f32(
                       false, a, false, b1, (short)0, acc1, false, false);
            acc2 = __builtin_amdgcn_wmma_f32_16x16x4_f32(
                       false, a, false, b2, (short)0, acc2, false, false);
        }
        __syncthreads();        // done reading before next chunk overwrites
    }

    // ---- log-softmax loss: VGPR j holds rows m=j (lanes 0-15), m=j+8 ----
    float wacc = 0.0f;
    #pragma unroll
    for (int j = 0; j < 8; ++j) {
        const int   m  = j + 8 * h;
        const float v0 = acc0[j];              // c = n
        const float v1 = acc1[j];              // c = 16 + n
        const float v2 = acc2[j];              // c = 32 (lane n==0 only)

        float lm = fmaxf(v0, v1);
        if (n == 0) lm = fmaxf(lm, v2);
        lm = fmaxf(lm, SWZ(lm, 0x041F));       // xor 1  (stays in 16-group)
        lm = fmaxf(lm, SWZ(lm, 0x081F));       // xor 2
        lm = fmaxf(lm, SWZ(lm, 0x101F));       // xor 4
        lm = fmaxf(lm, SWZ(lm, 0x201F));       // xor 8

        float es = __expf(v0 - lm) + __expf(v1 - lm) +
                   ((n == 0) ? __expf(v2 - lm) : 0.0f);
        es += SWZ(es, 0x041F);
        es += SWZ(es, 0x081F);
        es += SWZ(es, 0x101F);
        es += SWZ(es, 0x201F);

        if (n == 0 && m < STEPS && u >= m)     // valid rows only
            wacc += __logf(es) + lm - v0;
    }
    wacc += SWZ(wacc, 0x401F);                 // xor 16: combine halves
    if (lane == 0) partial[s * T_DIM + u] = wacc;
}

// Deterministic fixed-order reduction of the 16384 per-(s,u) partials.
__global__ __launch_bounds__(256) void cpc_reduce_kernel(
    const float* __restrict__ partial, float* __restrict__ out)
{
    __shared__ float sm[256];
    float a = 0.0f;
    for (int i = threadIdx.x; i < S_DIM * T_DIM; i += 256) a += partial[i];
    sm[threadIdx.x] = a;
    __syncthreads();
    #pragma unroll
    for (int stride = 128; stride > 0; stride >>= 1) {
        if ((int)threadIdx.x < stride) sm[threadIdx.x] += sm[threadIdx.x + stride];
        __syncthreads();
    }
    if (threadIdx.x == 0) out[0] = sm[0];
}

extern "C" void kernel_launch(void* const* d_in, const int* in_sizes, int n_in,
                              void* d_out, int out_size, void* d_ws, size_t ws_size,
                              hipStream_t stream) {
    const float* true_latent = (const float*)d_in[0];  // (32, 512, 512) f32
    const float* predictions = (const float*)d_in[1];  // (32, 512, 512, 12) f32
    const int*   neg_indices = (const int*)  d_in[2];  // (32, 32*512) i32
    float* partial = (float*)d_ws;                     // 32*512 floats = 64 KB

    const int nblocks = S_DIM * (T_DIM / U_TILE);      // 4096 blocks x 128 thr
    cpc_wmma_loss_kernel<<<nblocks, NTHREADS, 0, stream>>>(
        true_latent, predictions, neg_indices, partial);
    cpc_reduce_kernel<<<1, 256, 0, stream>>>(partial, (float*)d_out);
}